// GCN2_16887811408593
// MI455X (gfx1250) — compile-verified
//
#include <hip/hip_runtime.h>
#include <hip/hip_bf16.h>

typedef float v2f __attribute__((ext_vector_type(2)));
typedef float v8f __attribute__((ext_vector_type(8)));

// ---------------------------------------------------------------------------
// Zero-fill a float buffer
// ---------------------------------------------------------------------------
__global__ void gcn_zero_kernel(float* __restrict__ p, long long n) {
    long long i = (long long)blockIdx.x * blockDim.x + threadIdx.x;
    if (i < n) p[i] = 0.0f;
}

// ---------------------------------------------------------------------------
// In-degree via f32 atomics: deg[dst] += 1
// ---------------------------------------------------------------------------
__global__ void gcn_degree_kernel(const long long* __restrict__ ei,
                                  float* __restrict__ deg,
                                  long long n_edges) {
    long long e = (long long)blockIdx.x * blockDim.x + threadIdx.x;
    if (e >= n_edges) return;
    long long d = ei[n_edges + e];           // dst row of edge_index
    atomicAdd(&deg[d], 1.0f);
}

// ---------------------------------------------------------------------------
// dinv[i] = deg>0 ? rsqrt(max(deg,1)) : 0
// ---------------------------------------------------------------------------
__global__ void gcn_dinv_kernel(const float* __restrict__ deg,
                                float* __restrict__ dinv, int n) {
    int i = blockIdx.x * blockDim.x + threadIdx.x;
    if (i >= n) return;
    float d = deg[i];
    dinv[i] = (d > 0.0f) ? rsqrtf(fmaxf(d, 1.0f)) : 0.0f;
}

// ---------------------------------------------------------------------------
// FP32 GEMM via V_WMMA_F32_16X16X4_F32.
// C[M x NCOLS] = A[M x K] * B[K x NCOLS], row-major, K % 4 == 0, M % 16 == 0.
// One wave owns a 16-row strip and accumulates all NCOLS/16 column tiles,
// reusing the A fragment across tiles each K-step.
//
// A fragment (32-bit 16x4): lanes 0-15 -> M=lane, K = k0,k0+1 in v[0],v[1];
//                           lanes 16-31 -> same M, K = k0+2,k0+3.
// B fragment mirrors with N striped across lanes.
// C/D: VGPR r holds M=r (lanes 0-15) and M=r+8 (lanes 16-31), N = lane&15.
// ---------------------------------------------------------------------------
template <int K, int NCOLS>
__global__ void gcn_gemm_wmma_kernel(const float* __restrict__ A,
                                     const float* __restrict__ B,
                                     float* __restrict__ C,
                                     int m_tiles) {
    constexpr int NT = NCOLS / 16;
    const int wave = threadIdx.x >> 5;
    const int lane = threadIdx.x & 31;
    const int tileM = blockIdx.x * (blockDim.x >> 5) + wave;
    if (tileM >= m_tiles) return;            // wave-uniform guard: EXEC all-1s

    const int row = lane & 15;               // M (for A) / N (for B,C)
    const int hi  = lane >> 4;               // K-half selector
    const float* Arow = A + (size_t)(tileM * 16 + row) * K;

    v8f acc[NT];
#pragma unroll
    for (int t = 0; t < NT; ++t) acc[t] = v8f{0,0,0,0,0,0,0,0};

    for (int k0 = 0; k0 < K; k0 += 4) {
        v2f a;
        a.x = Arow[k0 + hi * 2 + 0];
        a.y = Arow[k0 + hi * 2 + 1];
#pragma unroll
        for (int t = 0; t < NT; ++t) {
            v2f b;
            b.x = B[(size_t)(k0 + hi * 2 + 0) * NCOLS + t * 16 + row];
            b.y = B[(size_t)(k0 + hi * 2 + 1) * NCOLS + t * 16 + row];
            acc[t] = __builtin_amdgcn_wmma_f32_16x16x4_f32(
                false, a, false, b, (short)0, acc[t], false, false);
        }
    }

#pragma unroll
    for (int t = 0; t < NT; ++t) {
        float* Cout = C + (size_t)(tileM * 16) * NCOLS + t * 16 + row;
#pragma unroll
        for (int r = 0; r < 8; ++r) {
            Cout[(size_t)(r + hi * 8) * NCOLS] = acc[t][r];
        }
    }
}

// ---------------------------------------------------------------------------
// Edge scatter, D=128: one wave per edge; lane carries 4 contiguous floats
// (float4 gather from h[src], 4x global_atomic_add_f32 into out[dst]).
// ---------------------------------------------------------------------------
__global__ void gcn_scatter128_kernel(const float* __restrict__ h,
                                      const long long* __restrict__ ei,
                                      const float* __restrict__ dinv,
                                      float* __restrict__ out,
                                      long long n_edges) {
    long long e = (long long)blockIdx.x * (blockDim.x >> 5) + (threadIdx.x >> 5);
    if (e >= n_edges) return;
    int lane = threadIdx.x & 31;
    long long s = ei[e];
    long long d = ei[n_edges + e];
    float nrm = dinv[s] * dinv[d];
    const float4 v = *(const float4*)(h + (size_t)s * 128 + lane * 4);
    float* o = out + (size_t)d * 128 + lane * 4;
    atomicAdd(o + 0, v.x * nrm);
    atomicAdd(o + 1, v.y * nrm);
    atomicAdd(o + 2, v.z * nrm);
    atomicAdd(o + 3, v.w * nrm);
}

// ---------------------------------------------------------------------------
// Edge scatter, D=64: one wave per edge; lane carries 2 contiguous floats.
// ---------------------------------------------------------------------------
__global__ void gcn_scatter64_kernel(const float* __restrict__ h,
                                     const long long* __restrict__ ei,
                                     const float* __restrict__ dinv,
                                     float* __restrict__ out,
                                     long long n_edges) {
    long long e = (long long)blockIdx.x * (blockDim.x >> 5) + (threadIdx.x >> 5);
    if (e >= n_edges) return;
    int lane = threadIdx.x & 31;
    long long s = ei[e];
    long long d = ei[n_edges + e];
    float nrm = dinv[s] * dinv[d];
    const float2 v = *(const float2*)(h + (size_t)s * 64 + lane * 2);
    float* o = out + (size_t)d * 64 + lane * 2;
    atomicAdd(o + 0, v.x * nrm);
    atomicAdd(o + 1, v.y * nrm);
}

// ---------------------------------------------------------------------------
// out[i] = relu(out[i] + b[i % D])  (RELU=1) or out[i] += b[i % D] (RELU=0)
// ---------------------------------------------------------------------------
template <int D, int RELU>
__global__ void gcn_bias_kernel(float* __restrict__ out,
                                const float* __restrict__ b,
                                long long n) {
    long long i = (long long)blockIdx.x * blockDim.x + threadIdx.x;
    if (i >= n) return;
    float v = out[i] + b[i % D];
    out[i] = RELU ? fmaxf(v, 0.0f) : v;
}

extern "C" void kernel_launch(void* const* d_in, const int* in_sizes, int n_in,
                              void* d_out, int out_size, void* d_ws, size_t ws_size,
                              hipStream_t stream) {
    const float*     x  = (const float*)d_in[0];      // [N,128]
    const long long* ei = (const long long*)d_in[1];  // [2,E] int64
    const float*     W1 = (const float*)d_in[2];      // [128,128]
    const float*     b1 = (const float*)d_in[3];      // [128]
    const float*     W2 = (const float*)d_in[4];      // [128,64]
    const float*     b2 = (const float*)d_in[5];      // [64]
    float*           out = (float*)d_out;             // [N,64]

    const long long N = in_sizes[0] / 128;            // 100000
    const long long E = in_sizes[1] / 2;              // 1600000

    // Workspace layout (floats): deg | dinv | h (N*128, reused for h2) | agg1 (N*128)
    float* deg  = (float*)d_ws;
    float* dinv = deg  + N;
    float* h    = dinv + N;            // byte offset 2N*4 = 800000, 16B aligned
    float* agg1 = h    + N * 128;      // 16B aligned

    const int T = 256;
    auto blk = [](long long n, int t) { return (unsigned)((n + t - 1) / t); };

    // 1) zero accumulators (every call: deterministic, harness doesn't re-poison)
    gcn_zero_kernel<<<blk(N, T), T, 0, stream>>>(deg, N);
    gcn_zero_kernel<<<blk(N * 128, T), T, 0, stream>>>(agg1, N * 128);
    gcn_zero_kernel<<<blk(N * 64, T), T, 0, stream>>>(out, N * 64);

    // 2) degrees + symmetric norm factors
    gcn_degree_kernel<<<blk(E, T), T, 0, stream>>>(ei, deg, E);
    gcn_dinv_kernel<<<blk(N, T), T, 0, stream>>>(deg, dinv, (int)N);

    // 3) layer 1: h = x @ W1 (WMMA f32)
    const int m_tiles = (int)(N / 16);                // 6250, exact
    const int waves_per_blk = T / 32;                 // 8
    gcn_gemm_wmma_kernel<128, 128>
        <<<blk(m_tiles, waves_per_blk), T, 0, stream>>>(x, W1, h, m_tiles);

    // 4) gather/scatter aggregation (one wave per edge)
    gcn_scatter128_kernel<<<blk(E, waves_per_blk), T, 0, stream>>>(h, ei, dinv, agg1, E);

    // 5) bias + relu in place
    gcn_bias_kernel<128, 1><<<blk(N * 128, T), T, 0, stream>>>(agg1, b1, N * 128);

    // 6) layer 2: h2 = relu_out @ W2 (reuse h buffer)
    gcn_gemm_wmma_kernel<128, 64>
        <<<blk(m_tiles, waves_per_blk), T, 0, stream>>>(agg1, W2, h, m_tiles);

    // 7) aggregate into output
    gcn_scatter64_kernel<<<blk(E, waves_per_blk), T, 0, stream>>>(h, ei, dinv, out, E);

    // 8) final bias (no relu)
    gcn_bias_kernel<64, 0><<<blk(N * 64, T), T, 0, stream>>>(out, b2, N * 64);
}